// AtomAttentionDecoder_317827580850
// MI455X (gfx1250) — compile-verified
//
#include <hip/hip_runtime.h>

// ---------------------------------------------------------------------------
// AtomAttentionDecoder on MI455X (gfx1250): fp32 WMMA (V_WMMA_F32_16X16X4_F32)
// B=2 N=16384 T=2048 D=128 S=384 NW=512 WQ=32 H=128 L=3 NH=4 DH=32 NTOK=33
// Layout tuned so every gemm_phase WMMA fragment is a single aligned ds_load_b64.
// ---------------------------------------------------------------------------

typedef float v2f __attribute__((ext_vector_type(2)));
typedef float v8f __attribute__((ext_vector_type(8)));

#define CB    2
#define CN    16384
#define CT    2048
#define CD    128
#define CS    384
#define CNW   512
#define CWQ   32
#define CH    128
#define CL    3
#define CNH   4
#define CDH   32
#define CNTOK 33
#define CR    (CB*CN)      /* 32768 rows */
#define CBT   (CB*CT)      /* 4096 token rows */

#define LAP   34           /* lA row pitch: even (8B-aligned pairs), 34m%64 distinct */
#define KTP   130          /* Kt row pitch: even, conflict-free */

__device__ __forceinline__ float sigm(float x) { return 1.0f / (1.0f + __expf(-x)); }

// Block-wide (128 threads) sum reduction through shared memory.
__device__ __forceinline__ float blk_sum128(float v, float* sh, int t) {
  sh[t] = v; __syncthreads();
  for (int s = 64; s > 0; s >>= 1) { if (t < s) sh[t] += sh[t + s]; __syncthreads(); }
  float r = sh[0]; __syncthreads();
  return r;
}

// ---------------------------------------------------------------------------
// Core WMMA GEMM phase: 256 threads (8 waves). Block computes 128 rows x 128
// cols of A(rows x K) @ W(K x 128-slice). Wave wv owns rows [wv*16, wv*16+16).
// LDS:
//   lA[128][LAP]  row-major A chunk; (a.x,a.y) = A[m][k..k+1] is an aligned v2f.
//   lW           32x128 W chunk stored pair-interleaved:
//                lW[(k/2)*256 + n*2 + (k&1)] = W[k][n]
//                so (B[k][n],B[k+1][n]) (k even) is one aligned v2f (b64 load).
// f32 16x16x4 fragment layout (ISA 7.12.2):
//   A: a.x=A[m][k0+2*half], a.y=A[m][k0+2*half+1], m=l16 (+16 per M-tile)
//   B: b.x=B[k0+2*half][n], b.y=B[k0+2*half+1][n], n=l16+16*t
//   D: D[r+8*half][l16+16*t] = acc[t][r]
// ---------------------------------------------------------------------------
__device__ void gemm_phase(const float* __restrict__ A, int lda, int K,
                           const float* __restrict__ W, int ldw,
                           float* lA, float* lW, v8f acc[8],
                           long rowbase, int tid, int wv, int l16, int hlf) {
  for (int k0 = 0; k0 < K; k0 += 32) {
    __syncthreads();
    for (int i = tid; i < 128 * 32; i += 256) {
      int r = i >> 5, c = i & 31;
      lA[r * LAP + c] = A[(rowbase + r) * (long)lda + k0 + c];
    }
    for (int i = tid; i < 32 * 128; i += 256) {
      int kr = i >> 7, c = i & 127;
      lW[(kr >> 1) * 256 + c * 2 + (kr & 1)] = W[(long)(k0 + kr) * ldw + c];
    }
    if (k0 + 32 < K) {  // prefetch next chunk (global_prefetch_b8)
      __builtin_prefetch(&A[(rowbase + (tid & 127)) * (long)lda + k0 + 32], 0, 3);
      __builtin_prefetch(&W[(long)(k0 + 32 + (tid & 31)) * ldw], 0, 3);
    }
    __syncthreads();
    const int m = wv * 16 + l16;
    for (int kc = 0; kc < 32; kc += 4) {
      const int kr0 = kc + 2 * hlf;                  // even
      v2f a = *(const v2f*)&lA[m * LAP + kr0];
      const float* lwp = &lW[(kr0 >> 1) * 256];
      for (int t = 0; t < 8; ++t) {
        v2f b = *(const v2f*)&lwp[(t * 16 + l16) * 2];
        acc[t] = __builtin_amdgcn_wmma_f32_16x16x4_f32(
            false, a, false, b, (short)0, acc[t], false, false);
      }
    }
  }
}

// ---------------------------------------------------------------------------
// Dual-GEMM with fused epilogue.
// EPI 0: out0 = lnq * sigmoid(A@W0) + A@W1                (hcur / h2)
// EPI 1: out0 = A@W0 ; out1 = sigmoid(A@W1)               (qh, gate)
// EPI 2: out0 = A@W0 ; out1 = A@W1                        (kproj, vproj)
// EPI 3: out0 = silu(A@W0) * (A@W1)                       (FF inner, 256 cols)
// EPI 4: out0 = A@W0                                      (token proj)
// ---------------------------------------------------------------------------
template <int EPI>
__global__ __launch_bounds__(256) void dual_gemm_kernel(
    const float* __restrict__ A, int lda, int K,
    const float* __restrict__ W0, const float* __restrict__ W1, int ldw,
    float* __restrict__ out0, float* __restrict__ out1, int ldo, int ocol,
    const float* __restrict__ aux) {
  __shared__ float lA[128 * LAP];
  __shared__ float lW[32 * 128];
  const int tid = threadIdx.x, wv = tid >> 5, lane = tid & 31;
  const int l16 = lane & 15, hlf = lane >> 4;
  const long rowbase = (long)blockIdx.x * 128;

  v8f acc0[8], acc1[8];
  for (int t = 0; t < 8; ++t)
    for (int i = 0; i < 8; ++i) { acc0[t][i] = 0.0f; acc1[t][i] = 0.0f; }

  gemm_phase(A, lda, K, W0, ldw, lA, lW, acc0, rowbase, tid, wv, l16, hlf);
  if (EPI != 4)
    gemm_phase(A, lda, K, W1, ldw, lA, lW, acc1, rowbase, tid, wv, l16, hlf);

  for (int t = 0; t < 8; ++t) {
    for (int r = 0; r < 8; ++r) {
      long row = rowbase + wv * 16 + r + 8 * hlf;
      long idx = row * (long)ldo + ocol + t * 16 + l16;
      float p0 = acc0[t][r], p1 = acc1[t][r];
      if (EPI == 0) {
        float ln = aux[row * CD + t * 16 + l16];
        out0[idx] = ln * sigm(p0) + p1;
      } else if (EPI == 1) {
        out0[idx] = p0;
        out1[idx] = sigm(p1);
      } else if (EPI == 2) {
        out0[idx] = p0;
        out1[idx] = p1;
      } else if (EPI == 3) {
        out0[idx] = (p0 * sigm(p0)) * p1;   // silu(p0) * p1
      } else {
        out0[idx] = p0;
      }
    }
  }
}

// ---------------------------------------------------------------------------
// Residual two-phase GEMM: q += sigmoid(A1@W1) * (A0@W0)
// (attention-output and FF-output residual branches)
// ---------------------------------------------------------------------------
__global__ __launch_bounds__(256) void residual_gemm_kernel(
    const float* __restrict__ A0, int K0, const float* __restrict__ W0,
    const float* __restrict__ A1, int K1, const float* __restrict__ W1,
    float* __restrict__ qbuf) {
  __shared__ float lA[128 * LAP];
  __shared__ float lW[32 * 128];
  const int tid = threadIdx.x, wv = tid >> 5, lane = tid & 31;
  const int l16 = lane & 15, hlf = lane >> 4;
  const long rowbase = (long)blockIdx.x * 128;

  v8f acc0[8], acc1[8];
  for (int t = 0; t < 8; ++t)
    for (int i = 0; i < 8; ++i) { acc0[t][i] = 0.0f; acc1[t][i] = 0.0f; }

  gemm_phase(A0, K0, K0, W0, CD, lA, lW, acc0, rowbase, tid, wv, l16, hlf);
  gemm_phase(A1, K1, K1, W1, CD, lA, lW, acc1, rowbase, tid, wv, l16, hlf);

  for (int t = 0; t < 8; ++t) {
    for (int r = 0; r < 8; ++r) {
      long row = rowbase + wv * 16 + r + 8 * hlf;
      long idx = row * (long)CD + t * 16 + l16;
      qbuf[idx] += sigm(acc1[t][r]) * acc0[t][r];
    }
  }
}

// ---------------------------------------------------------------------------
// q = q_in + gather(ta); cn = LayerNorm(c) (no affine). One block per row.
// ---------------------------------------------------------------------------
__global__ __launch_bounds__(128) void prep_kernel(
    const float* __restrict__ q_in, const float* __restrict__ c_in,
    const float* __restrict__ ta, float* __restrict__ qbuf,
    float* __restrict__ cn) {
  __shared__ float sh[128];
  const long row = blockIdx.x;
  const int b = (int)(row >> 14);           // / CN
  const long n = row & (CN - 1);
  const int t = threadIdx.x;
  float qv = q_in[row * CD + t] + ta[((long)b * CT + (n >> 3)) * CD + t];
  qbuf[row * CD + t] = qv;
  float cv = c_in[row * CD + t];
  float m = blk_sum128(cv, sh, t) * (1.0f / CD);
  float var = blk_sum128(cv * cv, sh, t) * (1.0f / CD) - m * m;
  cn[row * CD + t] = (cv - m) * rsqrtf(var + 1e-5f);
}

__global__ __launch_bounds__(128) void lnq_kernel(
    const float* __restrict__ qbuf, float* __restrict__ lnq) {
  __shared__ float sh[128];
  const long row = blockIdx.x;
  const int t = threadIdx.x;
  float v = qbuf[row * CD + t];
  float m = blk_sum128(v, sh, t) * (1.0f / CD);
  float var = blk_sum128(v * v, sh, t) * (1.0f / CD) - m * m;
  lnq[row * CD + t] = (v - m) * rsqrtf(var + 1e-5f);
}

// ---------------------------------------------------------------------------
// Windowed attention. Grid (NW, 2 head-pairs, B), block = 64 threads (2 waves,
// one head each). K window is a contiguous slab [keybase, keybase+128).
// Kt LDS holds K transposed (row pitch KTP=130, conflict-free, 8B-aligned
// pairs). Scores + attn@V via f32 WMMA; softmax on the register-resident
// D-matrix layout with shfl_xor; attn spilled into the Kt rows this wave
// alone consumed.
// ---------------------------------------------------------------------------
__global__ __launch_bounds__(64) void attn_kernel(
    const float* __restrict__ qh, const float* __restrict__ kproj,
    const float* __restrict__ vproj, const float* __restrict__ gbuf,
    const float* __restrict__ adb, const float* __restrict__ amask,
    const int* __restrict__ key_idx, float* __restrict__ obuf) {
  __shared__ float Kt[64 * KTP];
  const int w = blockIdx.x, hp = blockIdx.y, b = blockIdx.z;
  const int tid = threadIdx.x, wv = tid >> 5, lane = tid & 31;
  const int l16 = lane & 15, hlf = lane >> 4;
  const int h = hp * 2 + wv;
  const int keybase = key_idx[w * CH];      // windows are contiguous slabs

  // Stage K^T for this head pair: Kt[col_local][key]
  for (int key = 0; key < CH; ++key)
    Kt[tid * KTP + key] =
        kproj[((long)b * CN + keybase + key) * CD + hp * 64 + tid];
  __syncthreads();

  const int kl0 = wv * 32;                  // this wave's rows in Kt

  // ---- scores = Q . K^T  (32 q rows x 128 keys, K-dim = 32) ----
  v8f sacc[2][8];
  for (int mt = 0; mt < 2; ++mt)
    for (int nt = 0; nt < 8; ++nt)
      for (int i = 0; i < 8; ++i) sacc[mt][nt][i] = 0.0f;

  for (int mt = 0; mt < 2; ++mt) {
    const float* qp =
        qh + ((long)b * CN + w * CWQ + mt * 16 + l16) * CD + h * CDH;
    for (int kc = 0; kc < CDH; kc += 4) {
      const int kr0 = kc + 2 * hlf;
      v2f a;
      a.x = qp[kr0];
      a.y = qp[kr0 + 1];
      for (int nt = 0; nt < 8; ++nt) {
        v2f bb;
        bb.x = Kt[(kl0 + kr0) * KTP + nt * 16 + l16];
        bb.y = Kt[(kl0 + kr0 + 1) * KTP + nt * 16 + l16];
        sacc[mt][nt] = __builtin_amdgcn_wmma_f32_16x16x4_f32(
            false, a, false, bb, (short)0, sacc[mt][nt], false, false);
      }
    }
  }

  // ---- scale + decoder bias + mask bias ----
  const float scale = 0.1767766953f;        // 1/sqrt(32)
  for (int mt = 0; mt < 2; ++mt)
    for (int nt = 0; nt < 8; ++nt)
      for (int r = 0; r < 8; ++r) {
        int qr = mt * 16 + r + 8 * hlf;
        int key = nt * 16 + l16;
        float bias =
            adb[((((long)b * CNW + w) * CNH + h) * CWQ + qr) * CH + key];
        float msk = amask[(long)b * CN + key_idx[w * CH + key]];
        sacc[mt][nt][r] =
            sacc[mt][nt][r] * scale + bias + (1.0f - msk) * (-1e9f);
      }

  // ---- softmax over keys (per query row) ----
  for (int mt = 0; mt < 2; ++mt) {
    float mx[8], sm[8];
    for (int r = 0; r < 8; ++r) {
      mx[r] = sacc[mt][0][r];
      for (int nt = 1; nt < 8; ++nt) mx[r] = fmaxf(mx[r], sacc[mt][nt][r]);
    }
    for (int off = 1; off < 16; off <<= 1)
      for (int r = 0; r < 8; ++r)
        mx[r] = fmaxf(mx[r], __shfl_xor(mx[r], off, 32));
    for (int r = 0; r < 8; ++r) sm[r] = 0.0f;
    for (int nt = 0; nt < 8; ++nt)
      for (int r = 0; r < 8; ++r) {
        float e = __expf(sacc[mt][nt][r] - mx[r]);
        sacc[mt][nt][r] = e;
        sm[r] += e;
      }
    for (int off = 1; off < 16; off <<= 1)
      for (int r = 0; r < 8; ++r) sm[r] += __shfl_xor(sm[r], off, 32);
    for (int nt = 0; nt < 8; ++nt)
      for (int r = 0; r < 8; ++r) sacc[mt][nt][r] /= sm[r];
  }

  // Spill attn into the Kt rows this wave alone consumed (wave-local reuse).
  for (int mt = 0; mt < 2; ++mt)
    for (int nt = 0; nt < 8; ++nt)
      for (int r = 0; r < 8; ++r)
        Kt[(kl0 + mt * 16 + r + 8 * hlf) * KTP + nt * 16 + l16] =
            sacc[mt][nt][r];

  // ---- out = attn @ V   (32 x 32 per head, K-dim = 128 keys) ----
  v8f oacc[2][2];
  for (int mt = 0; mt < 2; ++mt)
    for (int nt = 0; nt < 2; ++nt)
      for (int i = 0; i < 8; ++i) oacc[mt][nt][i] = 0.0f;

  for (int mt = 0; mt < 2; ++mt)
    for (int kc = 0; kc < CH; kc += 4) {
      const int kr0 = kc + 2 * hlf;
      v2f a = *(const v2f*)&Kt[(kl0 + mt * 16 + l16) * KTP + kr0];
      for (int nt = 0; nt < 2; ++nt) {
        const float* vp = vproj + ((long)b * CN + keybase + kr0) * CD +
                          h * CDH + nt * 16 + l16;
        v2f bb;
        bb.x = vp[0];
        bb.y = vp[CD];
        oacc[mt][nt] = __builtin_amdgcn_wmma_f32_16x16x4_f32(
            false, a, false, bb, (short)0, oacc[mt][nt], false, false);
      }
    }

  // ---- gate with g = sigmoid(hcur@Wg) and store ----
  for (int mt = 0; mt < 2; ++mt)
    for (int nt = 0; nt < 2; ++nt)
      for (int r = 0; r < 8; ++r) {
        long row = (long)b * CN + w * CWQ + mt * 16 + r + 8 * hlf;
        long col = h * CDH + nt * 16 + l16;
        obuf[row * CD + col] = oacc[mt][nt][r] * gbuf[row * CD + col];
      }
}

// ---------------------------------------------------------------------------
// r_update = (ln(q)*g + b) @ Wpos  (3 output cols, scalar reduction kernel)
// ---------------------------------------------------------------------------
__global__ __launch_bounds__(128) void final_r_kernel(
    const float* __restrict__ qbuf, const float* __restrict__ ln_g,
    const float* __restrict__ ln_b, const float* __restrict__ Wpos,
    float* __restrict__ out) {
  __shared__ float sh[128];
  const long row = blockIdx.x;
  const int t = threadIdx.x;
  float v = qbuf[row * CD + t];
  float m = blk_sum128(v, sh, t) * (1.0f / CD);
  float var = blk_sum128(v * v, sh, t) * (1.0f / CD) - m * m;
  float y = (v - m) * rsqrtf(var + 1e-5f) * ln_g[t] + ln_b[t];
  for (int j = 0; j < 3; ++j) {
    float s = blk_sum128(y * Wpos[t * 3 + j], sh, t);
    if (t == 0) out[row * 3 + j] = s;
  }
}

// ---------------------------------------------------------------------------
// res_type = segment_sum(q*mask over 8 atoms/token) @ Wres + bres
// ---------------------------------------------------------------------------
__global__ __launch_bounds__(128) void final_res_kernel(
    const float* __restrict__ qbuf, const float* __restrict__ amask,
    const float* __restrict__ Wres, const float* __restrict__ bres,
    float* __restrict__ out) {
  __shared__ float sf[128];
  const long tok = blockIdx.x;
  const int b = (int)(tok >> 11);           // / CT
  const long tt = tok & (CT - 1);
  const int t = threadIdx.x;
  const long nbase = (long)b * CN + tt * 8;
  float s = 0.0f;
  for (int i = 0; i < 8; ++i)
    s += qbuf[(nbase + i) * CD + t] * amask[nbase + i];
  sf[t] = s;
  __syncthreads();
  if (t < CNTOK) {
    float acc = bres[t];
    for (int k = 0; k < CD; ++k) acc += sf[k] * Wres[k * CNTOK + t];
    out[tok * CNTOK + t] = acc;
  }
}

// ---------------------------------------------------------------------------
extern "C" void kernel_launch(void* const* d_in, const int* in_sizes, int n_in,
                              void* d_out, int out_size, void* d_ws,
                              size_t ws_size, hipStream_t stream) {
  (void)in_sizes; (void)n_in; (void)out_size; (void)ws_size;
  const float* a       = (const float*)d_in[0];
  const float* q_in    = (const float*)d_in[1];
  const float* c_in    = (const float*)d_in[2];
  const float* adb     = (const float*)d_in[3];
  /* d_in[4] atom_to_token: one-hot n -> n/8, applied analytically */
  const float* amask   = (const float*)d_in[5];
  const int*   key_idx = (const int*)d_in[6];
  const float* a2qw    = (const float*)d_in[7];
  const float* Wq   = (const float*)d_in[8];
  const float* Wk   = (const float*)d_in[9];
  const float* Wv   = (const float*)d_in[10];
  const float* Wg   = (const float*)d_in[11];
  const float* Wo   = (const float*)d_in[12];
  const float* Wcs  = (const float*)d_in[13];
  const float* Wcb  = (const float*)d_in[14];
  const float* Wog  = (const float*)d_in[15];
  const float* Wcs2 = (const float*)d_in[16];
  const float* Wcb2 = (const float*)d_in[17];
  const float* Wog2 = (const float*)d_in[18];
  const float* W1   = (const float*)d_in[19];
  const float* W2   = (const float*)d_in[20];
  const float* W3   = (const float*)d_in[21];
  const float* ln_g = (const float*)d_in[22];
  const float* ln_b = (const float*)d_in[23];
  const float* Wpos = (const float*)d_in[24];
  const float* Wres = (const float*)d_in[25];
  const float* bres = (const float*)d_in[26];
  float* out = (float*)d_out;

  // Workspace layout (floats): ~187 MB total, L2-resident.
  float* ws = (float*)d_ws;
  const size_t RD = (size_t)CR * CD;        // 4.19M floats
  float* cn    = ws;
  float* qbuf  = ws + 1 * RD;
  float* lnq   = ws + 2 * RD;
  float* hcur  = ws + 3 * RD;               // reused as h2
  float* qh    = ws + 4 * RD;
  float* gbuf  = ws + 5 * RD;
  float* kproj = ws + 6 * RD;
  float* vproj = ws + 7 * RD;
  float* obuf  = ws + 8 * RD;
  float* ubuf  = ws + 9 * RD;               // (R, 256)
  float* ta    = ws + 11 * RD;              // (B*T, 128)

  const int DD = CD * CD;
  dim3 gblk(CR / 128);

  // ta = a @ a_to_q_w  (rows = B*T, K = S)
  dual_gemm_kernel<4><<<dim3(CBT / 128), 256, 0, stream>>>(
      a, CS, CS, a2qw, nullptr, CD, ta, nullptr, CD, 0, nullptr);
  // q = q_in + gather(ta); cn = ln(c)
  prep_kernel<<<CR, 128, 0, stream>>>(q_in, c_in, ta, qbuf, cn);

  for (int l = 0; l < CL; ++l) {
    const float* Wq_l = Wq + l * DD;   const float* Wk_l = Wk + l * DD;
    const float* Wv_l = Wv + l * DD;   const float* Wg_l = Wg + l * DD;
    const float* Wo_l = Wo + l * DD;   const float* Wcs_l = Wcs + l * DD;
    const float* Wcb_l = Wcb + l * DD; const float* Wog_l = Wog + l * DD;
    const float* Wcs2_l = Wcs2 + l * DD; const float* Wcb2_l = Wcb2 + l * DD;
    const float* Wog2_l = Wog2 + l * DD;
    const float* W1_l = W1 + (size_t)l * CD * 2 * CD;
    const float* W2_l = W2 + (size_t)l * CD * 2 * CD;
    const float* W3_l = W3 + (size_t)l * 2 * CD * CD;

    lnq_kernel<<<CR, 128, 0, stream>>>(qbuf, lnq);
    // hcur = ln(q)*sigmoid(cn@Wcs) + cn@Wcb
    dual_gemm_kernel<0><<<gblk, 256, 0, stream>>>(
        cn, CD, CD, Wcs_l, Wcb_l, CD, hcur, nullptr, CD, 0, lnq);
    // qh = hcur@Wq ; g = sigmoid(hcur@Wg)
    dual_gemm_kernel<1><<<gblk, 256, 0, stream>>>(
        hcur, CD, CD, Wq_l, Wg_l, CD, qh, gbuf, CD, 0, nullptr);
    // kproj = hcur@Wk ; vproj = hcur@Wv (project all rows once, gather later)
    dual_gemm_kernel<2><<<gblk, 256, 0, stream>>>(
        hcur, CD, CD, Wk_l, Wv_l, CD, kproj, vproj, CD, 0, nullptr);
    // windowed attention -> obuf = g * (attn @ V)
    attn_kernel<<<dim3(CNW, 2, CB), 64, 0, stream>>>(
        qh, kproj, vproj, gbuf, adb, amask, key_idx, obuf);
    // q += sigmoid(cn@Wog) * (o@Wo)
    residual_gemm_kernel<<<gblk, 256, 0, stream>>>(
        obuf, CD, Wo_l, cn, CD, Wog_l, qbuf);

    lnq_kernel<<<CR, 128, 0, stream>>>(qbuf, lnq);
    // h2 = ln(q)*sigmoid(cn@Wcs2) + cn@Wcb2   (stored in hcur)
    dual_gemm_kernel<0><<<gblk, 256, 0, stream>>>(
        cn, CD, CD, Wcs2_l, Wcb2_l, CD, hcur, nullptr, CD, 0, lnq);
    // u = silu(h2@W1) * (h2@W2)   (256 cols, two column halves)
    for (int ch = 0; ch < 2; ++ch)
      dual_gemm_kernel<3><<<gblk, 256, 0, stream>>>(
          hcur, CD, CD, W1_l + ch * CD, W2_l + ch * CD, 2 * CD,
          ubuf, nullptr, 2 * CD, ch * CD, nullptr);
    // q += sigmoid(cn@Wog2) * (u@W3)
    residual_gemm_kernel<<<gblk, 256, 0, stream>>>(
        ubuf, 2 * CD, W3_l, cn, CD, Wog2_l, qbuf);
  }

  // r_update -> out[0 : R*3)
  final_r_kernel<<<CR, 128, 0, stream>>>(qbuf, ln_g, ln_b, Wpos, out);
  // res_type -> out[R*3 : R*3 + BT*33)
  final_res_kernel<<<CBT, 128, 0, stream>>>(qbuf, amask, Wres, bres,
                                            out + (size_t)CR * 3);
}